// MultiHeadGraphAttention_22960895165079
// MI455X (gfx1250) — compile-verified
//
#include <hip/hip_runtime.h>
#include <hip/hip_bf16.h>
#include <math.h>

#define N_NODES 50000
#define N_EDGES 800000
#define D_FEAT  128
#define HEADS   8
#define UNITS   32
#define CH      (HEADS * UNITS)   // 256
#define LEAKY   0.2f
#define EPS_F   1e-7f

typedef float v2f __attribute__((ext_vector_type(2)));
typedef float v8f __attribute__((ext_vector_type(8)));

// ---------------------------------------------------------------------------
// Kernel 1: H = X (50000x128) @ W (128x256), fp32 WMMA 16x16x4.
// One wave32 computes one 16x16 tile of H. 50000 is an exact multiple of 16,
// so every wave is fully active (EXEC all-ones as WMMA requires).
// ---------------------------------------------------------------------------
__global__ __launch_bounds__(256) void gat_gemm_wmma(
    const float* __restrict__ X, const float* __restrict__ W,
    float* __restrict__ H) {
  const int wave = (int)((blockIdx.x * blockDim.x + threadIdx.x) >> 5);
  const int lane = threadIdx.x & 31;
  const int tn = wave & 15;    // 16 tiles across N = 256
  const int tm = wave >> 4;    // 3125 tiles down M = 50000
  if (tm >= N_NODES / 16) return;   // never taken (exact grid), safety only

  const int l  = lane & 15;
  const int hi = lane >> 4;    // lanes 16-31 carry K+2 / K+3 per ISA layout

  const float* arow = X + (size_t)(tm * 16 + l) * D_FEAT;
  const float* bcol = W + tn * 16 + l;

  v8f acc = {0.f, 0.f, 0.f, 0.f, 0.f, 0.f, 0.f, 0.f};
#pragma unroll
  for (int k0 = 0; k0 < D_FEAT; k0 += 4) {
    const int ka = k0 + 2 * hi;
    v2f a; a.x = arow[ka];                 a.y = arow[ka + 1];
    v2f b; b.x = bcol[(size_t)ka * CH];    b.y = bcol[(size_t)(ka + 1) * CH];
    acc = __builtin_amdgcn_wmma_f32_16x16x4_f32(
        /*neg_a=*/false, a, /*neg_b=*/false, b,
        /*c_mod=*/(short)0, acc, /*reuse_a=*/false, /*reuse_b=*/false);
  }

  // C/D layout: VGPR v -> row (v + 8*hi), lanes l = column.
  float* orow = H + (size_t)(tm * 16 + 8 * hi) * CH + tn * 16 + l;
#pragma unroll
  for (int v = 0; v < 8; ++v) orow[(size_t)v * CH] = acc[v];
}

// ---------------------------------------------------------------------------
// Kernel 2: init accumulators (out = 0, segmax = -inf, denom = 0)
// ---------------------------------------------------------------------------
__global__ __launch_bounds__(256) void gat_init(
    float* __restrict__ out, float* __restrict__ smax, float* __restrict__ den) {
  const int idx = blockIdx.x * blockDim.x + threadIdx.x;
  if (idx < N_NODES * CH)   out[idx]  = 0.0f;
  if (idx < N_NODES * HEADS) { smax[idx] = -INFINITY; den[idx] = 0.0f; }
}

// ---------------------------------------------------------------------------
// Kernel 3: per-node head scores  g[n,h] = sum_u H[n,h,u] * ka1[h,u]
// ---------------------------------------------------------------------------
__global__ __launch_bounds__(256) void gat_node_scores(
    const float* __restrict__ H, const float* __restrict__ ka1,
    float* __restrict__ G) {
  const int idx = blockIdx.x * blockDim.x + threadIdx.x;   // n*HEADS + head
  if (idx >= N_NODES * HEADS) return;
  const int n = idx >> 3, head = idx & 7;
  const float* hp = H + (size_t)n * CH + head * UNITS;
  const float* ap = ka1 + head * UNITS;
  float s = 0.0f;
#pragma unroll
  for (int u = 0; u < UNITS; ++u) s = fmaf(hp[u], ap[u], s);
  G[idx] = s;
}

// ---------------------------------------------------------------------------
// Kernel 4: edge scores + segment max.
// Reference quirk preserved: f_s[e] = f_t[src[e]] = g[ tgt[ src[e] ] ].
// ---------------------------------------------------------------------------
__global__ __launch_bounds__(256) void gat_edge_scores(
    const int* __restrict__ edges, const float* __restrict__ G,
    float* __restrict__ S, float* __restrict__ smax) {
  const int idx = blockIdx.x * blockDim.x + threadIdx.x;   // e*HEADS + head
  if (idx >= N_EDGES * HEADS) return;
  const int e = idx >> 3, head = idx & 7;
  const int t  = edges[2 * e + 1];
  const int sn = edges[2 * e + 0];
  const int t2 = edges[2 * sn + 1];          // targets[sources[e]]
  float v = G[t * HEADS + head] + G[t2 * HEADS + head];
  v = (v > 0.0f) ? v : LEAKY * v;            // leaky_relu
  S[idx] = v;
  atomicMax(&smax[t * HEADS + head], v);     // global_atomic_max_num_f32
}

// ---------------------------------------------------------------------------
// Kernel 5: exp(score - segmax) + segment sum
// ---------------------------------------------------------------------------
__global__ __launch_bounds__(256) void gat_softmax_num(
    const int* __restrict__ edges, const float* __restrict__ smax,
    float* __restrict__ S, float* __restrict__ den) {
  const int idx = blockIdx.x * blockDim.x + threadIdx.x;
  if (idx >= N_EDGES * HEADS) return;
  const int e = idx >> 3, head = idx & 7;
  const int t = edges[2 * e + 1];
  const float v = expf(S[idx] - smax[t * HEADS + head]);
  S[idx] = v;
  atomicAdd(&den[t * HEADS + head], v);      // global_atomic_add_f32
}

// ---------------------------------------------------------------------------
// Kernel 6: weighted aggregation. One 256-thread block per edge:
// coalesced read of H[src] row, coalesced atomic-add burst into out[tgt] row.
// ---------------------------------------------------------------------------
__global__ __launch_bounds__(256) void gat_aggregate(
    const int* __restrict__ edges, const float* __restrict__ H,
    const float* __restrict__ S, const float* __restrict__ den,
    float* __restrict__ out) {
  const int e = blockIdx.x;
  const int c = threadIdx.x;                 // channel 0..255
  const int t  = edges[2 * e + 1];
  const int sn = edges[2 * e + 0];
  const int head = c >> 5;
  const float alpha = S[e * HEADS + head] / (den[t * HEADS + head] + EPS_F);
  atomicAdd(&out[(size_t)t * CH + c], alpha * H[(size_t)sn * CH + c]);
}

// ---------------------------------------------------------------------------
// Kernel 7: epilogue  out = elu(out + bias)
// ---------------------------------------------------------------------------
__global__ __launch_bounds__(256) void gat_elu(
    float* __restrict__ out, const float* __restrict__ bias) {
  const int idx = blockIdx.x * blockDim.x + threadIdx.x;
  if (idx >= N_NODES * CH) return;
  const float v = out[idx] + bias[idx & (CH - 1)];
  out[idx] = (v > 0.0f) ? v : expm1f(v);
}

// ---------------------------------------------------------------------------
extern "C" void kernel_launch(void* const* d_in, const int* in_sizes, int n_in,
                              void* d_out, int out_size, void* d_ws, size_t ws_size,
                              hipStream_t stream) {
  const float* x     = (const float*)d_in[0];   // (50000,128)
  const int*   edges = (const int*)  d_in[1];   // (800000,2)
  const float* w     = (const float*)d_in[2];   // (128,256)
  const float* ka1   = (const float*)d_in[3];   // (1,8,32)
  // d_in[4] = ka2 -- unused by the reference
  const float* bias  = (const float*)d_in[5];   // (256,)
  float* out = (float*)d_out;

  char* ws = (char*)d_ws;
  float* H    = (float*)(ws + 0);          // 50000*256*4 = 51,200,000 B
  float* G    = (float*)(ws + 51200000);   // 400,000*4   =  1,600,000 B
  float* S    = (float*)(ws + 52800000);   // 6,400,000*4 = 25,600,000 B
  float* SMAX = (float*)(ws + 78400000);   // 400,000*4   =  1,600,000 B
  float* DEN  = (float*)(ws + 80000000);   // 400,000*4   =  1,600,000 B

  const int threads = 256;

  // init accumulators (d_out and ws are poisoned by the harness)
  gat_init<<<(N_NODES * CH + threads - 1) / threads, threads, 0, stream>>>(out, SMAX, DEN);

  // GEMM: 50000 tiles * 32 lanes = 1,600,000 threads = 6250 blocks exactly
  gat_gemm_wmma<<<(N_NODES / 16) * 16 * 32 / threads, threads, 0, stream>>>(x, w, H);

  gat_node_scores<<<(N_NODES * HEADS + threads - 1) / threads, threads, 0, stream>>>(H, ka1, G);

  gat_edge_scores<<<(N_EDGES * HEADS + threads - 1) / threads, threads, 0, stream>>>(edges, G, S, SMAX);

  gat_softmax_num<<<(N_EDGES * HEADS + threads - 1) / threads, threads, 0, stream>>>(edges, SMAX, S, DEN);

  gat_aggregate<<<N_EDGES, threads, 0, stream>>>(edges, H, S, DEN, out);

  gat_elu<<<(N_NODES * CH + threads - 1) / threads, threads, 0, stream>>>(out, bias);
}